// processor_28114855920231
// MI455X (gfx1250) — compile-verified
//
#include <hip/hip_runtime.h>
#include <math.h>

// ---------------- problem constants (from reference) ----------------
#define Bb   64
#define NIN  256
#define OUTD 256
#define Mm   128
#define Nn   1024
#define Rr   4
#define Hh   512
#define FOURH 2048
__device__ __constant__ float kEPS = 1e-8f;

typedef float v2f __attribute__((ext_vector_type(2)));
typedef float v8f __attribute__((ext_vector_type(8)));

__device__ __forceinline__ float sigmoidf_(float x) { return 1.0f / (1.0f + expf(-x)); }
__device__ __forceinline__ float softplusf_(float x) { return fmaxf(x, 0.0f) + log1pf(expf(-fabsf(x))); }
__device__ __forceinline__ float apply_act(float v, int act) {
    if (act == 1) return sigmoidf_(v);
    if (act == 2) return 1.0f + softplusf_(v);
    return v;
}

// ---------------------------------------------------------------
// Generic fp32 linear via V_WMMA_F32_16X16X4_F32:
//   Y[m,n] = act( sum_p X_p[m,:] . W_p[n,:] + bias1[n] (+bias2[n]) )
// One wave computes one 16x16 tile of Y. Mrows fixed = 64 (B).
// ---------------------------------------------------------------
__global__ __launch_bounds__(128) void wmma_linear_kernel(
    float* __restrict__ Y, int ldY,
    const float* __restrict__ X1, int ld1, const float* __restrict__ W1, int ldw1, int K1,
    const float* __restrict__ X2, int ld2, const float* __restrict__ W2, int ldw2, int K2,
    const float* __restrict__ X3, int ld3, const float* __restrict__ W3, int ldw3, int K3,
    const float* __restrict__ bias1, const float* __restrict__ bias2,
    int Nout, int act)
{
    const int lane = threadIdx.x & 31;
    const int wid  = threadIdx.x >> 5;
    const int tile = blockIdx.x * 4 + wid;
    const int tiles = 4 * (Nout >> 4);     // (64/16) * (Nout/16)
    if (tile >= tiles) return;             // wave-uniform

    const int m0 = (tile & 3) * 16;
    const int n0 = (tile >> 2) * 16;
    const int half = lane >> 4;            // 0: K0/K1, 1: K2/K3
    const int lidx = lane & 15;

    v8f acc = {0.f,0.f,0.f,0.f,0.f,0.f,0.f,0.f};

    const float* xr;
    const float* wr;

#define K_LOOP(Xp, ldp, Wp, ldwp, Kp)                                              \
    if (Kp > 0) {                                                                  \
        xr = Xp + (size_t)(m0 + lidx) * (ldp);                                     \
        wr = Wp + (size_t)(n0 + lidx) * (ldwp);                                    \
        for (int k0 = 0; k0 < (Kp); k0 += 4) {                                     \
            v2f a = *(const v2f*)(xr + k0 + 2 * half);                             \
            v2f b = *(const v2f*)(wr + k0 + 2 * half);                             \
            acc = __builtin_amdgcn_wmma_f32_16x16x4_f32(false, a, false, b,        \
                                                        (short)0, acc, false, false); \
        }                                                                          \
    }

    K_LOOP(X1, ld1, W1, ldw1, K1)
    K_LOOP(X2, ld2, W2, ldw2, K2)
    K_LOOP(X3, ld3, W3, ldw3, K3)
#undef K_LOOP

    // D layout: VGPR v: lanes0-15 -> row m0+v, lanes16-31 -> row m0+8+v; col = n0+lidx
    const int col = n0 + lidx;
    float bv = (bias1 ? bias1[col] : 0.0f) + (bias2 ? bias2[col] : 0.0f);
    const int rbase = m0 + half * 8;
#pragma unroll
    for (int v = 0; v < 8; ++v) {
        float val = acc[v] + bv;
        Y[(size_t)(rbase + v) * ldY + col] = apply_act(val, act);
    }
}

// ---------------- small linear heads (tiny Nout) ----------------
__global__ void small_linear_kernel(float* __restrict__ Y,
                                    const float* __restrict__ X, int ldx,
                                    const float* __restrict__ W,
                                    const float* __restrict__ bias,
                                    int Mrows, int Nout, int K, int act)
{
    int gid = blockIdx.x * blockDim.x + threadIdx.x;
    if (gid >= Mrows * Nout) return;
    int m = gid / Nout, n = gid % Nout;
    float acc = bias ? bias[n] : 0.0f;
    const float* xr = X + (size_t)m * ldx;
    const float* wr = W + (size_t)n * K;
    for (int k = 0; k < K; ++k) acc += xr[k] * wr[k];
    Y[gid] = apply_act(acc, act);
}

// ---------------- LSTM elementwise ----------------
__global__ void lstm_kernel(const float* __restrict__ gates,
                            const float* __restrict__ prev_c,
                            float* __restrict__ h)
{
    int gid = blockIdx.x * blockDim.x + threadIdx.x;
    if (gid >= Bb * Hh) return;
    int b = gid / Hh, j = gid % Hh;
    const float* g = gates + (size_t)b * FOURH;
    float ig = g[j], fg = g[Hh + j], gg = g[2 * Hh + j], og = g[3 * Hh + j];
    float c = sigmoidf_(fg) * prev_c[gid] + sigmoidf_(ig) * tanhf(gg);
    h[gid] = sigmoidf_(og) * tanhf(c);
}

// ---------------- usage -> sort -> allocation (one block per batch) ----------------
__global__ __launch_bounds__(256) void alloc_kernel(
    const float* __restrict__ free_g,      // (B,R) sigmoid applied
    const float* __restrict__ prev_read_w, // (B,R,N)
    const float* __restrict__ prev_write_w,// (B,1,N)
    const float* __restrict__ prev_usage,  // (B,1,N)
    float* __restrict__ alloc)             // (B,N)
{
    __shared__ float key[Nn];
    __shared__ int   idx[Nn];
    __shared__ float cp[Nn];
    const int b = blockIdx.x, tid = threadIdx.x;

    float fgv[Rr];
#pragma unroll
    for (int r = 0; r < Rr; ++r) fgv[r] = free_g[b * Rr + r];

#pragma unroll
    for (int q = 0; q < 4; ++q) {
        int n = q * 256 + tid;
        float psi = 1.0f;
#pragma unroll
        for (int r = 0; r < Rr; ++r)
            psi *= (1.0f - fgv[r] * prev_read_w[((size_t)(b * Rr + r)) * Nn + n]);
        float pu = prev_usage[(size_t)b * Nn + n];
        float pw = prev_write_w[(size_t)b * Nn + n];
        key[n] = (pu + pw - pu * pw) * psi;
        idx[n] = n;
    }
    __syncthreads();

    // bitonic sort ascending, stable via index tie-break
    for (int k = 2; k <= Nn; k <<= 1) {
        for (int j = k >> 1; j > 0; j >>= 1) {
            for (int t = tid; t < Nn; t += 256) {
                int p = t ^ j;
                if (p > t) {
                    bool up = ((t & k) == 0);
                    float a = key[t], c = key[p];
                    int ia = idx[t], ic = idx[p];
                    bool sw = up ? (a > c || (a == c && ia > ic))
                                 : (a < c || (a == c && ia < ic));
                    if (sw) { key[t] = c; key[p] = a; idx[t] = ic; idx[p] = ia; }
                }
            }
            __syncthreads();
        }
    }

    // inclusive product scan (Hillis-Steele)
    for (int q = 0; q < 4; ++q) cp[q * 256 + tid] = key[q * 256 + tid];
    __syncthreads();
    for (int off = 1; off < Nn; off <<= 1) {
        float t4[4];
#pragma unroll
        for (int q = 0; q < 4; ++q) {
            int i = q * 256 + tid;
            t4[q] = (i >= off) ? cp[i] * cp[i - off] : cp[i];
        }
        __syncthreads();
#pragma unroll
        for (int q = 0; q < 4; ++q) cp[q * 256 + tid] = t4[q];
        __syncthreads();
    }

#pragma unroll
    for (int q = 0; q < 4; ++q) {
        int i = q * 256 + tid;
        alloc[(size_t)b * Nn + idx[i]] = (1.0f - key[i]) * cp[i];
    }
}

// ---------------- write content weighting + write_w (block per batch) ----------------
__global__ __launch_bounds__(256) void write_w_kernel(
    const float* __restrict__ memory,   // (B,N,M) original
    const float* __restrict__ wk,       // (B,M)
    const float* __restrict__ wbeta,    // (B)
    const float* __restrict__ alloc,    // (B,N)
    const float* __restrict__ ag,       // (B)
    const float* __restrict__ wg,       // (B)
    float* __restrict__ write_w)        // (B,N)
{
    __shared__ float keyv[Mm];
    __shared__ float sarr[Nn];
    __shared__ float red[256];
    const int b = blockIdx.x, tid = threadIdx.x;

    if (tid < Mm) keyv[tid] = wk[b * Mm + tid];
    __syncthreads();

    float ksq = 0.0f;
    for (int m = 0; m < Mm; ++m) ksq += keyv[m] * keyv[m];
    const float kn = sqrtf(ksq) + kEPS;
    const float beta = wbeta[b];

#pragma unroll
    for (int q = 0; q < 4; ++q) {
        int n = q * 256 + tid;
        const float4* row4 = (const float4*)(memory + ((size_t)b * Nn + n) * Mm);
        float dot = 0.0f, nn = 0.0f;
        for (int m4 = 0; m4 < Mm / 4; ++m4) {
            float4 v4 = row4[m4];
            const float* v = (const float*)&v4;
#pragma unroll
            for (int j = 0; j < 4; ++j) { dot += v[j] * keyv[4 * m4 + j]; nn += v[j] * v[j]; }
        }
        sarr[n] = beta * dot / (kn * (sqrtf(nn) + kEPS));
    }
    __syncthreads();

    // softmax over N
    float lm = -1e30f;
    for (int q = 0; q < 4; ++q) lm = fmaxf(lm, sarr[q * 256 + tid]);
    red[tid] = lm; __syncthreads();
    for (int s = 128; s > 0; s >>= 1) { if (tid < s) red[tid] = fmaxf(red[tid], red[tid + s]); __syncthreads(); }
    float maxv = red[0]; __syncthreads();
    float ls = 0.0f;
    for (int q = 0; q < 4; ++q) ls += expf(sarr[q * 256 + tid] - maxv);
    red[tid] = ls; __syncthreads();
    for (int s = 128; s > 0; s >>= 1) { if (tid < s) red[tid] += red[tid + s]; __syncthreads(); }
    float sumv = red[0]; __syncthreads();

    float agv = ag[b], wgv = wg[b];
#pragma unroll
    for (int q = 0; q < 4; ++q) {
        int n = q * 256 + tid;
        float cw = expf(sarr[n] - maxv) / sumv;
        write_w[(size_t)b * Nn + n] = wgv * (agv * alloc[(size_t)b * Nn + n] + (1.0f - agv) * cw);
    }
}

// ---------------- fused link-matrix streaming: fwd & bwd ----------------
// L[n,k] = (1 - ww[n] - ww[k])*prev_links[n,k] + ww[n]*prec[k];  L[n,n] = 0
// fwd[r,k] = sum_n rw[r,n]*L[n,k]   (partial per block -> atomics)
// bwd[r,n] = sum_k rw[r,k]*L[n,k]   (full per block -> direct store)
// Each thread owns 4 CONTIGUOUS k values -> one global_load_b128 per row.
__global__ __launch_bounds__(256) void links_kernel(
    const float* __restrict__ prev_links, // (B,N,N)
    const float* __restrict__ prev_prec,  // (B,N)
    const float* __restrict__ prev_read_w,// (B,R,N)
    const float* __restrict__ write_w,    // (B,N)
    float* __restrict__ fwd,              // (B,R,N)  pre-zeroed
    float* __restrict__ bwd)              // (B,R,N)
{
    const int b = blockIdx.x >> 4;
    const int n0 = (blockIdx.x & 15) * 64;
    const int tid = threadIdx.x;
    const int lane = tid & 31, wid = tid >> 5;
    const int kb = tid << 2;               // base k for this thread (4 contiguous)

    __shared__ float wwn[64];
    __shared__ float rwn[Rr][64];
    __shared__ float bred[64][8][Rr];

    if (tid < 64) wwn[tid] = write_w[(size_t)b * Nn + n0 + tid];
    { int r = tid >> 6, n = tid & 63;
      rwn[r][n] = prev_read_w[((size_t)(b * Rr + r)) * Nn + n0 + n]; }

    float4 wwk4 = *(const float4*)(write_w + (size_t)b * Nn + kb);
    float4 pk4  = *(const float4*)(prev_prec + (size_t)b * Nn + kb);
    float4 rwk4[Rr];
#pragma unroll
    for (int r = 0; r < Rr; ++r)
        rwk4[r] = *(const float4*)(prev_read_w + ((size_t)(b * Rr + r)) * Nn + kb);
    const float* wwk = (const float*)&wwk4;
    const float* pk  = (const float*)&pk4;

    float facc[Rr][4] = {};
    __syncthreads();

    for (int nl = 0; nl < 64; ++nl) {
        const int n = n0 + nl;
        const float wn = wwn[nl];
        const float rn0 = rwn[0][nl], rn1 = rwn[1][nl], rn2 = rwn[2][nl], rn3 = rwn[3][nl];
        float4 pl4 = *(const float4*)(prev_links + ((size_t)b * Nn + n) * Nn + kb);
        const float* pl = (const float*)&pl4;
        float bp0 = 0.f, bp1 = 0.f, bp2 = 0.f, bp3 = 0.f;
#pragma unroll
        for (int q = 0; q < 4; ++q) {
            float L = (1.0f - wn - wwk[q]) * pl[q] + wn * pk[q];
            L = (kb + q == n) ? 0.0f : L;
            facc[0][q] += rn0 * L; facc[1][q] += rn1 * L;
            facc[2][q] += rn2 * L; facc[3][q] += rn3 * L;
            bp0 += ((const float*)&rwk4[0])[q] * L;
            bp1 += ((const float*)&rwk4[1])[q] * L;
            bp2 += ((const float*)&rwk4[2])[q] * L;
            bp3 += ((const float*)&rwk4[3])[q] * L;
        }
#pragma unroll
        for (int off = 16; off > 0; off >>= 1) {
            bp0 += __shfl_xor(bp0, off, 32);
            bp1 += __shfl_xor(bp1, off, 32);
            bp2 += __shfl_xor(bp2, off, 32);
            bp3 += __shfl_xor(bp3, off, 32);
        }
        if (lane == 0) {
            bred[nl][wid][0] = bp0; bred[nl][wid][1] = bp1;
            bred[nl][wid][2] = bp2; bred[nl][wid][3] = bp3;
        }
    }
    __syncthreads();

    { int nl = tid >> 2, r = tid & 3;
      float s = 0.f;
#pragma unroll
      for (int w = 0; w < 8; ++w) s += bred[nl][w][r];
      bwd[((size_t)(b * Rr + r)) * Nn + n0 + nl] = s; }

#pragma unroll
    for (int r = 0; r < Rr; ++r)
#pragma unroll
        for (int q = 0; q < 4; ++q)
            atomicAdd(fwd + ((size_t)(b * Rr + r)) * Nn + kb + q, facc[r][q]);
}

// ---------------- read content + read_w (block per (b,r)) ----------------
__global__ __launch_bounds__(256) void read_w_kernel(
    const float* __restrict__ memory,   // (B,N,M) original
    const float* __restrict__ write_w,  // (B,N)
    const float* __restrict__ ev,       // (B,M) sigmoid applied
    const float* __restrict__ wv,       // (B,M)
    const float* __restrict__ rk,       // (B,R*M)
    const float* __restrict__ rbeta,    // (B,R)
    const float* __restrict__ rm_raw,   // (B,R,3)
    const float* __restrict__ fwd,      // (B,R,N)
    const float* __restrict__ bwd,      // (B,R,N)
    float* __restrict__ read_w)         // (B,R,N)
{
    __shared__ float keyv[Mm], evs[Mm], wvs[Mm];
    __shared__ float sarr[Nn];
    __shared__ float red[256];
    const int b = blockIdx.x >> 2, r = blockIdx.x & 3, tid = threadIdx.x;

    if (tid < Mm) {
        keyv[tid] = rk[(size_t)b * (Rr * Mm) + r * Mm + tid];
        evs[tid]  = ev[b * Mm + tid];
        wvs[tid]  = wv[b * Mm + tid];
    }
    __syncthreads();

    float ksq = 0.0f;
    for (int m = 0; m < Mm; ++m) ksq += keyv[m] * keyv[m];
    const float kn = sqrtf(ksq) + kEPS;
    const float beta = rbeta[b * Rr + r];

#pragma unroll
    for (int q = 0; q < 4; ++q) {
        int n = q * 256 + tid;
        const float4* row4 = (const float4*)(memory + ((size_t)b * Nn + n) * Mm);
        float wwv = write_w[(size_t)b * Nn + n];
        float dot = 0.0f, nn = 0.0f;
        for (int m4 = 0; m4 < Mm / 4; ++m4) {
            float4 v4 = row4[m4];
            const float* vc = (const float*)&v4;
#pragma unroll
            for (int j = 0; j < 4; ++j) {
                int m = 4 * m4 + j;
                float v = vc[j] * (1.0f - wwv * evs[m]) + wwv * wvs[m]; // updated memory
                dot += v * keyv[m]; nn += v * v;
            }
        }
        sarr[n] = beta * dot / (kn * (sqrtf(nn) + kEPS));
    }
    __syncthreads();

    float lm = -1e30f;
    for (int q = 0; q < 4; ++q) lm = fmaxf(lm, sarr[q * 256 + tid]);
    red[tid] = lm; __syncthreads();
    for (int s = 128; s > 0; s >>= 1) { if (tid < s) red[tid] = fmaxf(red[tid], red[tid + s]); __syncthreads(); }
    float maxv = red[0]; __syncthreads();
    float ls = 0.0f;
    for (int q = 0; q < 4; ++q) ls += expf(sarr[q * 256 + tid] - maxv);
    red[tid] = ls; __syncthreads();
    for (int s = 128; s > 0; s >>= 1) { if (tid < s) red[tid] += red[tid + s]; __syncthreads(); }
    float sumv = red[0];

    // read_mode softmax (3 values)
    float x0 = rm_raw[(b * Rr + r) * 3 + 0];
    float x1 = rm_raw[(b * Rr + r) * 3 + 1];
    float x2 = rm_raw[(b * Rr + r) * 3 + 2];
    float mx = fmaxf(x0, fmaxf(x1, x2));
    float e0 = expf(x0 - mx), e1 = expf(x1 - mx), e2 = expf(x2 - mx);
    float es = e0 + e1 + e2;
    float rm0 = e0 / es, rm1 = e1 / es, rm2 = e2 / es;

#pragma unroll
    for (int q = 0; q < 4; ++q) {
        int n = q * 256 + tid;
        float cw = expf(sarr[n] - maxv) / sumv;
        size_t o = ((size_t)(b * Rr + r)) * Nn + n;
        read_w[o] = rm0 * cw + rm1 * fwd[o] + rm2 * bwd[o];
    }
}

// ---------------- reads = read_w @ updated_mem ----------------
__global__ void reads_kernel(const float* __restrict__ memory,
                             const float* __restrict__ write_w,
                             const float* __restrict__ ev,
                             const float* __restrict__ wv,
                             const float* __restrict__ read_w,
                             float* __restrict__ reads) // (B,R,M)
{
    int gid = blockIdx.x * blockDim.x + threadIdx.x;
    if (gid >= Bb * Rr * Mm) return;
    int m = gid & (Mm - 1);
    int r = (gid >> 7) & 3;
    int b = gid >> 9;
    float evm = ev[b * Mm + m], wvm = wv[b * Mm + m];
    float acc = 0.0f;
    const float* rw = read_w + ((size_t)(b * Rr + r)) * Nn;
    const float* ww = write_w + (size_t)b * Nn;
    const float* mem = memory + (size_t)b * Nn * Mm + m;
    for (int n = 0; n < Nn; ++n) {
        float wwv = ww[n];
        float v = mem[(size_t)n * Mm] * (1.0f - wwv * evm) + wwv * wvm;
        acc += rw[n] * v;
    }
    reads[gid] = acc;
}

// =====================================================================
extern "C" void kernel_launch(void* const* d_in, const int* in_sizes, int n_in,
                              void* d_out, int out_size, void* d_ws, size_t ws_size,
                              hipStream_t stream) {
    (void)in_sizes; (void)n_in; (void)out_size; (void)ws_size;
    const float* X           = (const float*)d_in[0];
    const float* memory      = (const float*)d_in[1];
    const float* prev_read   = (const float*)d_in[2];
    const float* prev_h      = (const float*)d_in[3];
    const float* prev_c      = (const float*)d_in[4];
    const float* prev_read_w = (const float*)d_in[5];
    const float* prev_write_w= (const float*)d_in[6];
    const float* prev_usage  = (const float*)d_in[7];
    const float* prev_prec   = (const float*)d_in[8];
    const float* prev_links  = (const float*)d_in[9];
    const float* w_ih  = (const float*)d_in[10];
    const float* w_hh  = (const float*)d_in[11];
    const float* b_ih  = (const float*)d_in[12];
    const float* b_hh  = (const float*)d_in[13];
    const float* proc_w= (const float*)d_in[14];
    const float* proc_b= (const float*)d_in[15];
    const float* wv_w  = (const float*)d_in[16];
    const float* wv_b  = (const float*)d_in[17];
    const float* ev_w  = (const float*)d_in[18];
    const float* ev_b  = (const float*)d_in[19];
    const float* fg_w  = (const float*)d_in[20];
    const float* fg_b  = (const float*)d_in[21];
    const float* ag_w  = (const float*)d_in[22];
    const float* ag_b  = (const float*)d_in[23];
    const float* wg_w  = (const float*)d_in[24];
    const float* wg_b  = (const float*)d_in[25];
    const float* rm_w  = (const float*)d_in[26];
    const float* rm_b  = (const float*)d_in[27];
    const float* wk_w  = (const float*)d_in[28];
    const float* wk_b  = (const float*)d_in[29];
    const float* ws_w  = (const float*)d_in[30];
    const float* ws_b  = (const float*)d_in[31];
    const float* rk_w  = (const float*)d_in[32];
    const float* rk_b  = (const float*)d_in[33];
    const float* rs_w  = (const float*)d_in[34];
    const float* rs_b  = (const float*)d_in[35];
    float* out = (float*)d_out;

    // ---- workspace layout (floats) ----
    float* ws = (float*)d_ws;
    size_t off = 0;
    float* g_gates  = ws + off; off += (size_t)Bb * FOURH;
    float* g_h      = ws + off; off += (size_t)Bb * Hh;
    float* g_wv     = ws + off; off += (size_t)Bb * Mm;
    float* g_ev     = ws + off; off += (size_t)Bb * Mm;
    float* g_fg     = ws + off; off += (size_t)Bb * Rr;
    float* g_ag     = ws + off; off += (size_t)Bb;
    float* g_wg     = ws + off; off += (size_t)Bb;
    float* g_rm     = ws + off; off += (size_t)Bb * Rr * 3;
    float* g_wk     = ws + off; off += (size_t)Bb * Mm;
    float* g_wbeta  = ws + off; off += (size_t)Bb;
    float* g_rk     = ws + off; off += (size_t)Bb * Rr * Mm;
    float* g_rbeta  = ws + off; off += (size_t)Bb * Rr;
    float* g_alloc  = ws + off; off += (size_t)Bb * Nn;
    float* g_ww     = ws + off; off += (size_t)Bb * Nn;
    float* g_fwd    = ws + off; off += (size_t)Bb * Rr * Nn;
    float* g_bwd    = ws + off; off += (size_t)Bb * Rr * Nn;
    float* g_rdw    = ws + off; off += (size_t)Bb * Rr * Nn;
    float* g_reads  = ws + off; off += (size_t)Bb * Rr * Mm;

    // 1) LSTM gates = [X | prev_read] @ w_ih^T + prev_h @ w_hh^T + b_ih + b_hh
    wmma_linear_kernel<<<(4 * (FOURH / 16)) / 4, 128, 0, stream>>>(
        g_gates, FOURH,
        X, NIN, w_ih, NIN + Rr * Mm, NIN,
        prev_read, Rr * Mm, w_ih + NIN, NIN + Rr * Mm, Rr * Mm,
        prev_h, Hh, w_hh, Hh, Hh,
        b_ih, b_hh, FOURH, 0);

    // 2) LSTM elementwise -> h
    lstm_kernel<<<(Bb * Hh) / 256, 256, 0, stream>>>(g_gates, prev_c, g_h);

    // 3) big heads via WMMA
    wmma_linear_kernel<<<(4 * (Mm / 16)) / 4, 128, 0, stream>>>(
        g_wv, Mm, g_h, Hh, wv_w, Hh, Hh,
        nullptr, 0, nullptr, 0, 0, nullptr, 0, nullptr, 0, 0,
        wv_b, nullptr, Mm, 0);
    wmma_linear_kernel<<<(4 * (Mm / 16)) / 4, 128, 0, stream>>>(
        g_ev, Mm, g_h, Hh, ev_w, Hh, Hh,
        nullptr, 0, nullptr, 0, 0, nullptr, 0, nullptr, 0, 0,
        ev_b, nullptr, Mm, 1);
    wmma_linear_kernel<<<(4 * (Mm / 16)) / 4, 128, 0, stream>>>(
        g_wk, Mm, g_h, Hh, wk_w, Hh, Hh,
        nullptr, 0, nullptr, 0, 0, nullptr, 0, nullptr, 0, 0,
        wk_b, nullptr, Mm, 0);
    wmma_linear_kernel<<<(4 * ((Rr * Mm) / 16)) / 4, 128, 0, stream>>>(
        g_rk, Rr * Mm, g_h, Hh, rk_w, Hh, Hh,
        nullptr, 0, nullptr, 0, 0, nullptr, 0, nullptr, 0, 0,
        rk_b, nullptr, Rr * Mm, 0);

    // 4) tiny heads
    small_linear_kernel<<<(Bb * Rr + 255) / 256, 256, 0, stream>>>(g_fg, g_h, Hh, fg_w, fg_b, Bb, Rr, Hh, 1);
    small_linear_kernel<<<1, 256, 0, stream>>>(g_ag, g_h, Hh, ag_w, ag_b, Bb, 1, Hh, 1);
    small_linear_kernel<<<1, 256, 0, stream>>>(g_wg, g_h, Hh, wg_w, wg_b, Bb, 1, Hh, 1);
    small_linear_kernel<<<(Bb * Rr * 3 + 255) / 256, 256, 0, stream>>>(g_rm, g_h, Hh, rm_w, rm_b, Bb, Rr * 3, Hh, 0);
    small_linear_kernel<<<1, 256, 0, stream>>>(g_wbeta, g_h, Hh, ws_w, ws_b, Bb, 1, Hh, 2);
    small_linear_kernel<<<(Bb * Rr + 255) / 256, 256, 0, stream>>>(g_rbeta, g_h, Hh, rs_w, rs_b, Bb, Rr, Hh, 2);

    // 5) usage -> sort -> allocation
    alloc_kernel<<<Bb, 256, 0, stream>>>(g_fg, prev_read_w, prev_write_w, prev_usage, g_alloc);

    // 6) write content weighting + write_w
    write_w_kernel<<<Bb, 256, 0, stream>>>(memory, g_wk, g_wbeta, g_alloc, g_ag, g_wg, g_ww);

    // 7) fused link-matrix streaming pass (the 268 MB bandwidth bound)
    hipMemsetAsync(g_fwd, 0, (size_t)Bb * Rr * Nn * sizeof(float), stream);
    links_kernel<<<Bb * 16, 256, 0, stream>>>(prev_links, prev_prec, prev_read_w, g_ww, g_fwd, g_bwd);

    // 8) read content weighting + read_w
    read_w_kernel<<<Bb * Rr, 256, 0, stream>>>(memory, g_ww, g_ev, g_wv, g_rk, g_rbeta, g_rm,
                                               g_fwd, g_bwd, g_rdw);

    // 9) reads
    reads_kernel<<<(Bb * Rr * Mm) / 256, 256, 0, stream>>>(memory, g_ww, g_ev, g_wv, g_rdw, g_reads);

    // 10) out = [h | reads] @ proc_w^T + proc_b
    wmma_linear_kernel<<<(4 * (OUTD / 16)) / 4, 128, 0, stream>>>(
        out, OUTD,
        g_h, Hh, proc_w, Hh + Rr * Mm, Hh,
        g_reads, Rr * Mm, proc_w + Hh, Hh + Rr * Mm, Rr * Mm,
        nullptr, 0, nullptr, 0, 0,
        proc_b, nullptr, OUTD, 0);
}